// TestWrapperModule_7232724927034
// MI455X (gfx1250) — compile-verified
//
#include <hip/hip_runtime.h>
#include <stdint.h>

// Problem constants (match reference)
#define NROWS   20000
#define DIM     512
#define NNZ     5000
#define RPG     4                         // rows per group
#define TPB     256
#define KCHUNK  20                        // ceil(NNZ / TPB); sorted-M run merging per chunk
#define NGROUPS (NROWS / RPG)             // 5000, exact
#define GRPELEM (RPG * DIM)               // 2048 floats = 8192 B per array per group

static_assert(NROWS % RPG == 0, "exact groups");
static_assert(GRPELEM * 4 == TPB * 16 * 2, "2 b128 async loads per thread per array");

__device__ __forceinline__ uint32_t lds_addr(const void* p) {
    // flat shared-aperture address: low 32 bits == wave-relative LDS byte address
    return (uint32_t)(uintptr_t)p;
}

// CDNA5 async DMA: global -> LDS, 16B per lane, tracked by ASYNCcnt
__device__ __forceinline__ void async_b128(uint32_t lds, uint32_t voff, const void* base) {
    asm volatile("global_load_async_to_lds_b128 %0, %1, %2"
                 :: "v"(lds), "v"(voff), "s"(base)
                 : "memory");
}

__global__ __launch_bounds__(TPB) void sparse_tp_kernel(
        const float* __restrict__ x, const float* __restrict__ y,
        const float* __restrict__ scale,
        const int* __restrict__ M, const int* __restrict__ M1, const int* __restrict__ M2,
        float* __restrict__ out)
{
    __shared__ uint2 paths_s[NNZ];                    // {M1|M2<<9|M<<18, scale bits} : 40 KB
    __shared__ __align__(16) float xbuf[2][GRPELEM];  // 16 KB (double buffered)
    __shared__ __align__(16) float ybuf[2][GRPELEM];  // 16 KB
    __shared__ __align__(16) float outs[GRPELEM];     // 8 KB accumulator tile

    const int tid = threadIdx.x;

    // Stage packed path table in LDS once per block (source is 80 KB total, L2-hot).
    for (int i = tid; i < NNZ; i += TPB) {
        uint32_t p = (uint32_t)M1[i] | ((uint32_t)M2[i] << 9) | ((uint32_t)M[i] << 18);
        paths_s[i] = make_uint2(p, __float_as_uint(scale[i]));
    }
    for (int j = tid; j < GRPELEM; j += TPB) outs[j] = 0.0f;

    const uint32_t xl[2] = { lds_addr(&xbuf[0][0]), lds_addr(&xbuf[1][0]) };
    const uint32_t yl[2] = { lds_addr(&ybuf[0][0]), lds_addr(&ybuf[1][0]) };
    const uint32_t o0 = (uint32_t)tid * 16u;
    const uint32_t o1 = o0 + TPB * 16u;

    int g = blockIdx.x;
    const int stride = gridDim.x;
    int buf = 0;

    if (g < NGROUPS) {   // prologue: DMA group g into buffer 0
        const float* xb = x + (size_t)g * GRPELEM;
        const float* yb = y + (size_t)g * GRPELEM;
        async_b128(xl[0] + o0, o0, xb);
        async_b128(xl[0] + o1, o1, xb);
        async_b128(yl[0] + o0, o0, yb);
        async_b128(yl[0] + o1, o1, yb);
    }

    for (; g < NGROUPS; g += stride) {
        // Prefetch next group into the other buffer (clamped duplicate at tail: harmless,
        // keeps per-wave ASYNCcnt issuance uniform so the wait threshold is static).
        int gn = g + stride; if (gn >= NGROUPS) gn = g;
        {
            const int nb = buf ^ 1;
            const float* xb = x + (size_t)gn * GRPELEM;
            const float* yb = y + (size_t)gn * GRPELEM;
            async_b128(xl[nb] + o0, o0, xb);
            async_b128(xl[nb] + o1, o1, xb);
            async_b128(yl[nb] + o0, o0, yb);
            async_b128(yl[nb] + o1, o1, yb);
        }
        // Async loads complete in order: <=4 outstanding => current buffer is resident.
        asm volatile("s_wait_asynccnt 0x4" ::: "memory");
        __syncthreads();

        const float* xs = &xbuf[buf][0];
        const float* ys = &ybuf[buf][0];

        // Contiguous chunk of sorted paths -> register run-merge, rare LDS atomics.
        int i0 = tid * KCHUNK;
        int i1 = i0 + KCHUNK; if (i1 > NNZ) i1 = NNZ;
        float a0 = 0.f, a1 = 0.f, a2 = 0.f, a3 = 0.f;
        int cur = -1;
        for (int i = i0; i < i1; ++i) {
            const uint2 p = paths_s[i];
            const uint32_t m1 = p.x & 511u;
            const uint32_t m2 = (p.x >> 9) & 511u;
            const int      m  = (int)(p.x >> 18);
            const float    s  = __uint_as_float(p.y);
            if (m != cur) {
                if (cur >= 0) {
                    atomicAdd(&outs[cur],          a0);
                    atomicAdd(&outs[DIM + cur],    a1);
                    atomicAdd(&outs[2*DIM + cur],  a2);
                    atomicAdd(&outs[3*DIM + cur],  a3);
                }
                cur = m; a0 = a1 = a2 = a3 = 0.f;
            }
            a0 += s * xs[m1]         * ys[m2];
            a1 += s * xs[DIM + m1]   * ys[DIM + m2];
            a2 += s * xs[2*DIM + m1] * ys[2*DIM + m2];
            a3 += s * xs[3*DIM + m1] * ys[3*DIM + m2];
        }
        if (cur >= 0) {
            atomicAdd(&outs[cur],          a0);
            atomicAdd(&outs[DIM + cur],    a1);
            atomicAdd(&outs[2*DIM + cur],  a2);
            atomicAdd(&outs[3*DIM + cur],  a3);
        }
        __syncthreads();

        // Coalesced b128 stores; each thread re-zeros exactly what it stored.
        float4*       out4 = (float4*)out + (size_t)g * (GRPELEM / 4);
        float4*       os4  = (float4*)outs;
        const int e = tid * 2;
        out4[e]     = os4[e];
        out4[e + 1] = os4[e + 1];
        os4[e]      = make_float4(0.f, 0.f, 0.f, 0.f);
        os4[e + 1]  = make_float4(0.f, 0.f, 0.f, 0.f);

        buf ^= 1;
    }
}

extern "C" void kernel_launch(void* const* d_in, const int* in_sizes, int n_in,
                              void* d_out, int out_size, void* d_ws, size_t ws_size,
                              hipStream_t stream) {
    (void)in_sizes; (void)n_in; (void)d_ws; (void)ws_size; (void)out_size;
    const float* x     = (const float*)d_in[0];
    const float* y     = (const float*)d_in[1];
    const float* scale = (const float*)d_in[2];
    const int*   M     = (const int*)d_in[3];
    const int*   M1    = (const int*)d_in[4];
    const int*   M2    = (const int*)d_in[5];
    float* out = (float*)d_out;

    // Grid-stride over 5000 row-groups; 2048 blocks amortizes the 40 KB path-table
    // staging ~2.5x while still filling every WGP at 4 blocks/WGP (80 KB LDS each).
    sparse_tp_kernel<<<2048, TPB, 0, stream>>>(x, y, scale, M, M1, M2, out);
}